// AttnPool_20134806684031
// MI455X (gfx1250) — compile-verified
//
#include <hip/hip_runtime.h>

typedef float v2f __attribute__((ext_vector_type(2)));
typedef float v8f __attribute__((ext_vector_type(8)));

#define DIM 256
#define TILES_PER_WAVE 4            // 4 tiles x 16 rows = 64 rows per wave
#define ROWS_PER_WAVE (TILES_PER_WAVE * 16)

// ---------------------------------------------------------------------------
// Kernel 0: zero the output accumulator and the Z scalar in workspace.
// ---------------------------------------------------------------------------
__global__ void attnpool_init(float* __restrict__ out, int out_size,
                              float* __restrict__ Z) {
  int i = blockIdx.x * blockDim.x + threadIdx.x;
  if (i < out_size) out[i] = 0.0f;
  if (i == 0) *Z = 0.0f;
}

// ---------------------------------------------------------------------------
// Kernel 1 (fused single pass over x):
//   per 16-row tile: s = x_tile . w via V_WMMA_F32_16X16X4_F32 (f32 WMMA),
//   p = exp(s), then weighted scatter-accumulate p[r]*x[r,:] into per-segment
//   register accumulators (batch_index is sorted -> flush on segment change
//   with global f32 atomics). Also accumulates Z = sum(p) via block reduce.
//
//   Tail handling: out-of-range rows have their A-row index CLAMPED to N-1
//   (branch-free loads; EXEC stays all-ones around WMMA). The resulting
//   garbage scores are never consumed: the weighting loop breaks at row>=N
//   and zsum only accumulates valid rows.
// ---------------------------------------------------------------------------
__global__ __launch_bounds__(256) void attnpool_main(
    const float* __restrict__ x, const float* __restrict__ w,
    const int* __restrict__ bidx, float* __restrict__ out,
    float* __restrict__ Z, int N) {
  __shared__ float zbuf[8];

  const int lane = threadIdx.x & 31;
  const int wid  = threadIdx.x >> 5;
  const long gw  = (long)blockIdx.x * 8 + wid;   // global wave id
  const int mrow = lane & 15;                    // A-matrix row this lane owns
  const int half = lane >> 4;                    // K pair select (K=0,1 vs 2,3)

  float acc[8];
#pragma unroll
  for (int j = 0; j < 8; ++j) acc[j] = 0.0f;
  int cur_seg = -1;
  float zsum = 0.0f;

  for (int t = 0; t < TILES_PER_WAVE; ++t) {
    const long base = gw * ROWS_PER_WAVE + (long)t * 16;

    // ---- prefetch next tile's row for this lane (global_prefetch_b8) ----
    const long prow = base + 16 + mrow;
    if (prow < N) __builtin_prefetch(x + prow * DIM, 0, 1);

    // ---- f32 WMMA mat-vec: D[m,n] = sum_k x[base+m,k] * w[k] (all n) ----
    // Clamp the row so loads are unconditional (no exec-mask churn; scores of
    // clamped rows are dead values).
    long arow = base + mrow;
    if (arow >= (long)N) arow = (long)N - 1;
    const float* aptr = x + arow * DIM + 2 * half;
    const float* bptr = w + 2 * half;

    v8f c = {};
#pragma unroll 8
    for (int kb = 0; kb < DIM; kb += 4) {
      v2f a = *(const v2f*)(aptr + kb);           // A: 16x4 f32 fragment
      v2f b = *(const v2f*)(bptr + kb);           // B: w replicated over 16 cols
      c = __builtin_amdgcn_wmma_f32_16x16x4_f32(
          /*neg_a=*/false, a, /*neg_b=*/false, b,
          /*c_mod=*/(short)0, c, /*reuse_a=*/false, /*reuse_b=*/false);
    }

    // c[j] = s_{base + j + 8*half}  (replicated across the 16 N-columns)
    float pmine[8];
#pragma unroll
    for (int j = 0; j < 8; ++j) pmine[j] = __expf(c[j]);

    // exchange halves so every lane has exp(s) for all 16 rows of the tile
    float pw[16];
#pragma unroll
    for (int j = 0; j < 8; ++j) {
      float other = __shfl_xor(pmine[j], 16, 32);
      pw[j]     = (half == 0) ? pmine[j] : other;
      pw[8 + j] = (half == 0) ? other : pmine[j];
    }

    // ---- weighted scatter-accumulate: lane owns columns [lane*8, lane*8+8)
    for (int r = 0; r < 16; ++r) {
      const long row = base + r;
      if (row >= (long)N) break;                  // uniform across wave
      const int seg = bidx[row];                  // sorted
      if (seg != cur_seg) {
        if (cur_seg >= 0) {
          float* op = out + (long)cur_seg * DIM + lane * 8;
#pragma unroll
          for (int j = 0; j < 8; ++j) { atomicAdd(op + j, acc[j]); acc[j] = 0.0f; }
        }
        cur_seg = seg;
      }
      zsum += pw[r];  // pw[] is replicated across all lanes; lane 0's copy is used
      const float4* xp = (const float4*)(x + row * DIM + lane * 8);
      const float4 x0 = xp[0];
      const float4 x1 = xp[1];
      const float wr = pw[r];
      acc[0] += wr * x0.x; acc[1] += wr * x0.y;
      acc[2] += wr * x0.z; acc[3] += wr * x0.w;
      acc[4] += wr * x1.x; acc[5] += wr * x1.y;
      acc[6] += wr * x1.z; acc[7] += wr * x1.w;
    }
  }

  // final flush of the open segment
  if (cur_seg >= 0) {
    float* op = out + (long)cur_seg * DIM + lane * 8;
#pragma unroll
    for (int j = 0; j < 8; ++j) atomicAdd(op + j, acc[j]);
  }

  // ---- Z reduction: lane 0 of each wave -> LDS -> one atomic per block ----
  if (lane == 0) zbuf[wid] = zsum;
  __syncthreads();
  if (threadIdx.x == 0) {
    float s = 0.0f;
#pragma unroll
    for (int i = 0; i < 8; ++i) s += zbuf[i];
    atomicAdd(Z, s);
  }
}

// ---------------------------------------------------------------------------
// Kernel 2: out *= 1/Z  (softmax normalization is linear; applied at the end)
// ---------------------------------------------------------------------------
__global__ void attnpool_scale(float* __restrict__ out, int out_size,
                               const float* __restrict__ Z) {
  const float inv = 1.0f / *Z;
  int i = blockIdx.x * blockDim.x + threadIdx.x;
  if (i < out_size) out[i] *= inv;
}

// ---------------------------------------------------------------------------
extern "C" void kernel_launch(void* const* d_in, const int* in_sizes, int n_in,
                              void* d_out, int out_size, void* d_ws,
                              size_t ws_size, hipStream_t stream) {
  const float* x    = (const float*)d_in[0];
  const float* w    = (const float*)d_in[1];
  const int*   bidx = (const int*)d_in[2];
  // d_in[3] is the scalar B; out_size already encodes B*DIM.

  float* out = (float*)d_out;
  float* Z   = (float*)d_ws;

  const int N = in_sizes[0] / DIM;

  const int init_blocks = (out_size + 255) / 256;
  attnpool_init<<<init_blocks, 256, 0, stream>>>(out, out_size, Z);

  const int waves  = (N + ROWS_PER_WAVE - 1) / ROWS_PER_WAVE;
  const int blocks = (waves + 7) / 8;
  attnpool_main<<<blocks, 256, 0, stream>>>(x, w, bidx, out, Z, N);

  attnpool_scale<<<init_blocks, 256, 0, stream>>>(out, out_size, Z);
}